// Net_36971078484364
// MI455X (gfx1250) — compile-verified
//
#include <hip/hip_runtime.h>
#include <hip/hip_bf16.h>

// ---------------- problem constants (from reference) ----------------
#define KARY 16
#define NN   69905        // total nodes
#define EE   1000         // edges
#define DIM  64
#define VECD 300
#define INTERNAL_END 4369 // nodes [0,4369) are internal (have 16 children)
#define INV_DIV (1.0f / 17.0f)

typedef __attribute__((ext_vector_type(2))) float v2f;
typedef __attribute__((ext_vector_type(8))) float v8f;

// low 32 bits of a generic pointer to __shared__ = LDS byte offset
__device__ __forceinline__ unsigned lds_offset(const void* p) {
  return (unsigned)(unsigned long long)p;
}

// =====================================================================
// Phase A: base[i] = data_vecs[data[i]] @ data_W + data_b  -> emb[i]
// WMMA f32 16x16x4: one wave handles a tile of 16 nodes x 64 dims.
//   A (16x4, M=node, K=vec-dim): lane l (h=l>>4, m=l&15) holds
//       VGPR0 = A[m][kb+2h], VGPR1 = A[m][kb+2h+1]   (one b64 load)
//   B (4x16): lane l holds {B[2h][n], B[2h+1][n]}, n=l&15 — one ds_load_b64
//       from the K-pair-interleaved LDS image of data_W
//   C/D (16x16): lane l col n=l&15, VGPR v row m = v + 8h
// data_W is staged global->LDS with GLOBAL_LOAD_ASYNC_TO_LDS_B128
// (ASYNCcnt), then interleaved in-LDS into pair layout.
// =====================================================================
__global__ void base_wmma_kernel(const int* __restrict__ data,
                                 const float* __restrict__ data_vecs,
                                 const float* __restrict__ data_W,
                                 const float* __restrict__ data_b,
                                 float* __restrict__ emb) {
  extern __shared__ float smem[];
  float* sRaw  = smem;                  // [VECD][DIM]      raw copy (76.8 KB)
  float* sPair = smem + VECD * DIM;     // [VECD/2][2*DIM]  K-pair interleave

  const int tid = threadIdx.x;

  // ---- async stage of data_W into LDS (b128 per lane-transfer) ----
  {
    const float4* g4 = (const float4*)data_W;
    for (int i = tid; i < (VECD * DIM) / 4; i += blockDim.x) {
      unsigned loff = lds_offset(sRaw + i * 4);
      const float4* gp = g4 + i;
      asm volatile("global_load_async_to_lds_b128 %0, %1, off"
                   :: "v"(loff), "v"(gp) : "memory");
    }
    asm volatile("s_wait_asynccnt 0x0" ::: "memory");
  }
  __syncthreads();

  // ---- in-LDS interleave: sPair[kp][2n+j] = W[2kp+j][n] ----
  for (int idx = tid; idx < (VECD / 2) * DIM; idx += blockDim.x) {
    const int kp = idx >> 6;            // 0..149
    const int n  = idx & 63;
    v2f pr = { sRaw[(2 * kp) * DIM + n], sRaw[(2 * kp + 1) * DIM + n] };
    *(v2f*)(sPair + kp * (2 * DIM) + 2 * n) = pr;
  }
  __syncthreads();

  const int wave = tid >> 5;
  const int lane = tid & 31;
  const int ntiles = (NN + 15) >> 4;    // 4370
  const int tile = blockIdx.x * (blockDim.x >> 5) + wave;
  if (tile >= ntiles) return;           // wave-uniform

  const int m = lane & 15;              // A row / C col index
  const int h = lane >> 4;              // half-wave select
  int node = tile * 16 + m;
  int nodec = node < NN ? node : NN - 1;  // clamp (lanes stay converged)
  const long long row = (long long)data[nodec] * VECD;
  const float* aptr = data_vecs + row + 2 * h;

  v8f acc0 = {}, acc1 = {}, acc2 = {}, acc3 = {};

  for (int kb = 0; kb < VECD; kb += 4) {
    v2f a = *(const v2f*)(aptr + kb);
    // pair row index: (kb + 2h)/2 ; element pair n indexed directly
    const v2f* prow = (const v2f*)(sPair + ((kb >> 1) + h) * (2 * DIM)) + m;
    v2f bA = prow[0];
    v2f bB = prow[16];
    v2f bC = prow[32];
    v2f bD = prow[48];
    acc0 = __builtin_amdgcn_wmma_f32_16x16x4_f32(false, a, false, bA, (short)0, acc0, false, false);
    acc1 = __builtin_amdgcn_wmma_f32_16x16x4_f32(false, a, false, bB, (short)0, acc1, false, false);
    acc2 = __builtin_amdgcn_wmma_f32_16x16x4_f32(false, a, false, bC, (short)0, acc2, false, false);
    acc3 = __builtin_amdgcn_wmma_f32_16x16x4_f32(false, a, false, bD, (short)0, acc3, false, false);
  }

  // write out: emb[node, n] = acc + data_b[n]
  const int nbase = tile * 16;
#pragma unroll
  for (int t = 0; t < 4; ++t) {
    const int n = t * 16 + m;
    const float bb = data_b[n];
    v8f acc = t == 0 ? acc0 : (t == 1 ? acc1 : (t == 2 ? acc2 : acc3));
#pragma unroll
    for (int v = 0; v < 8; ++v) {
      const int nd = nbase + v + 8 * h;
      if (nd < NN) emb[(size_t)nd * DIM + n] = acc[v] + bb;
    }
  }
}

// =====================================================================
// Phase B: one level of bottom-up accumulation. One wave per parent p:
//   emb[p] += sum_{c=0..15} ( finals(16p+1+c) @ edge_W[edges[child]]
//                             + edge_b[edges[child]] )
// finals(x) = internal ? relu(emb[x]/17) : emb[x]   (leaf = raw base)
// =====================================================================
__global__ void level_kernel(const int* __restrict__ edges,
                             const float* __restrict__ edge_W,
                             const float* __restrict__ edge_b,
                             float* __restrict__ emb,
                             int parent_base, int nparents) {
  __shared__ float fin[8 * DIM];
  const int tid = threadIdx.x;
  const int wave = tid >> 5;
  const int lane = tid & 31;
  const int pidx = blockIdx.x * (blockDim.x >> 5) + wave;
  if (pidx >= nparents) return;               // wave-uniform
  const int p = parent_base + pidx;

  float* fs = fin + wave * DIM;
  const int k0 = 2 * lane;
  float acc0 = 0.f, acc1 = 0.f;

  for (int c = 0; c < KARY; ++c) {
    const int node = KARY * p + 1 + c;
    float v0 = emb[(size_t)node * DIM + lane];
    float v1 = emb[(size_t)node * DIM + lane + 32];
    if (node < INTERNAL_END) {                // uniform per wave
      v0 = fmaxf(v0 * INV_DIV, 0.f);
      v1 = fmaxf(v1 * INV_DIV, 0.f);
    }
    fs[lane] = v0;
    fs[lane + 32] = v1;
    __builtin_amdgcn_wave_barrier();

    const size_t e = (size_t)edges[node];
    const float* W = edge_W + e * (DIM * DIM) + k0;
    __builtin_prefetch(W, 0, 1);
#pragma unroll 8
    for (int d = 0; d < DIM; ++d) {
      const float f = fs[d];
      v2f w = *(const v2f*)(W + d * DIM);
      acc0 = fmaf(f, w.x, acc0);
      acc1 = fmaf(f, w.y, acc1);
    }
    acc0 += edge_b[e * DIM + k0];
    acc1 += edge_b[e * DIM + k0 + 1];
  }
  emb[(size_t)p * DIM + k0]     += acc0;
  emb[(size_t)p * DIM + k0 + 1] += acc1;
}

// =====================================================================
// Phase C: root = relu(emb[0]/17);
//   scores[e] = (root @ edge_W[e] + edge_b[e]) @ score_W   -> d_out[e]
// One wave per edge; wave32 shuffle reduction for the final dot.
// =====================================================================
__global__ void score_kernel(const float* __restrict__ edge_W,
                             const float* __restrict__ edge_b,
                             const float* __restrict__ score_W,
                             const float* __restrict__ emb,
                             float* __restrict__ out) {
  __shared__ float root[DIM];
  const int tid = threadIdx.x;
  if (tid < DIM) root[tid] = fmaxf(emb[tid] * INV_DIV, 0.f);
  __syncthreads();

  const int wave = tid >> 5;
  const int lane = tid & 31;
  const int e = blockIdx.x * (blockDim.x >> 5) + wave;
  if (e >= EE) return;

  const int k0 = 2 * lane;
  const float* W = edge_W + (size_t)e * (DIM * DIM) + k0;
  float acc0 = 0.f, acc1 = 0.f;
#pragma unroll 8
  for (int d = 0; d < DIM; ++d) {
    const float f = root[d];
    v2f w = *(const v2f*)(W + d * DIM);
    acc0 = fmaf(f, w.x, acc0);
    acc1 = fmaf(f, w.y, acc1);
  }
  acc0 += edge_b[(size_t)e * DIM + k0];
  acc1 += edge_b[(size_t)e * DIM + k0 + 1];

  float part = acc0 * score_W[k0] + acc1 * score_W[k0 + 1];
#pragma unroll
  for (int off = 16; off > 0; off >>= 1)
    part += __shfl_xor(part, off, 32);
  if (lane == 0) out[e] = part;
}

// =====================================================================
extern "C" void kernel_launch(void* const* d_in, const int* in_sizes, int n_in,
                              void* d_out, int out_size, void* d_ws, size_t ws_size,
                              hipStream_t stream) {
  (void)in_sizes; (void)n_in; (void)out_size; (void)ws_size;

  const int*   data      = (const int*)  d_in[0];
  /* d_in[1] = parents: unused, tree structure is closed-form */
  const int*   edges     = (const int*)  d_in[2];
  const float* data_vecs = (const float*)d_in[3];
  const float* data_W    = (const float*)d_in[4];
  const float* data_b    = (const float*)d_in[5];
  const float* edge_W    = (const float*)d_in[6];
  const float* edge_b    = (const float*)d_in[7];
  const float* score_W   = (const float*)d_in[8];
  float*       out       = (float*)d_out;
  float*       emb       = (float*)d_ws;   // N*DIM floats (~17.9 MB)

  // Phase A: base embeddings via WMMA (emb fully rewritten every call)
  const int ntiles  = (NN + 15) / 16;              // 4370
  const int blocksA = (ntiles + 7) / 8;            // 8 waves/block
  const size_t shA = (size_t)(2 * VECD * DIM) * sizeof(float);  // 153.6 KB
  base_wmma_kernel<<<blocksA, 256, shA, stream>>>(
      data, data_vecs, data_W, data_b, emb);

  // Phase B: levels 4 -> 1  (parent ranges of the complete 16-ary tree)
  // level l parents: base = (16^(l-1)-1)/15, count = 16^(l-1)
  const int pbase[4] = {273, 17, 1, 0};
  const int pcnt [4] = {4096, 256, 16, 1};
  for (int i = 0; i < 4; ++i) {
    const int blocks = (pcnt[i] + 7) / 8;
    level_kernel<<<blocks, 256, 0, stream>>>(edges, edge_W, edge_b, emb,
                                             pbase[i], pcnt[i]);
  }

  // Phase C: root scoring over all edges
  score_kernel<<<(EE + 7) / 8, 256, 0, stream>>>(edge_W, edge_b, score_W,
                                                 emb, out);
}